// MaximumAxisLoss_90580860272868
// MI455X (gfx1250) — compile-verified
//
#include <hip/hip_runtime.h>
#include <hip/hip_bf16.h>
#include <math.h>

typedef __attribute__((ext_vector_type(2))) float v2f;
typedef __attribute__((ext_vector_type(8))) float v8f;

#define B_      8
#define T_      100
#define N_      100000
#define ROWS    800          // B_*T_
#define ROW_TILES 50         // ROWS/16
#define COL_TILES 6250       // N_/16
#define COL_PAIRS 3125       // COL_TILES/2
#define MARGIN_ 0.1f
#define THRESH_ 0.5f
#define BIG_    1000000000.0f
#define HUGE_   3.0e38f

// ---------------- workspace layout (bytes) ----------------
// rowData : 800 * 16            [0      , 12800)
// bounds  : 48 floats (lv|uv)   [12800  , 12992)
// bcodes  : 100 uints           [12992  , 13392)
// gmin    : 800 uints           [13392  , 16592)
// pk      : 100000 * 32         [16592  , 3216592)

// ---------------------------------------------------------------------------
// Kernel 1: per-(b,t) row data (retraj xyz + r2), per-b bounds, b-codes, init
// ---------------------------------------------------------------------------
__global__ __launch_bounds__(1024)
void prep_rows(const float* __restrict__ outputs,   // (B,T,3)
               const float* __restrict__ c2ws,      // (B,4,4)
               const float* __restrict__ sscales,   // (B,)
               float*       __restrict__ rowData,   // 800*4 : x,y,z,r2
               float*       __restrict__ bounds,    // lv[24] | uv[24]
               unsigned*    __restrict__ bcodes,    // 100
               unsigned*    __restrict__ gmin)      // 800 ordered-uint keys
{
    int tid = threadIdx.x;
    if (tid < ROWS) {
        int b = tid / T_;
        float s = sscales[b];
        const float* M = c2ws + b * 16;           // row-major 4x4
        const float* o = outputs + tid * 3;
        float o0 = o[0], o1 = o[1], o2 = o[2];
        float4 r;
        r.x = (o0 * M[0] + o1 * M[1] + o2 * M[2])  * s + M[3];
        r.y = (o0 * M[4] + o1 * M[5] + o2 * M[6])  * s + M[7];
        r.z = (o0 * M[8] + o1 * M[9] + o2 * M[10]) * s + M[11];
        r.w = r.x * r.x + r.y * r.y + r.z * r.z;   // r2
        ((float4*)rowData)[tid] = r;
        gmin[tid] = 0x80000000u | __float_as_uint(BIG_);   // key(+BIG)
    }
    if (tid < ROW_TILES * 2) {                    // 4-bit b index per row, 8 rows/code
        int rt = tid >> 1, half = tid & 1;
        unsigned bc = 0;
        #pragma unroll
        for (int i = 0; i < 8; ++i) {
            unsigned bb = (unsigned)((rt * 16 + half * 8 + i) / T_);
            bc |= bb << (4 * i);
        }
        bcodes[tid] = bc;
    }
    __threadfence_block();
    __syncthreads();
    if (tid < 24) {                               // per-(b,dim) min/max over T
        int b = tid / 3, d = tid % 3;
        float mn = 3.0e38f, mx = -3.0e38f;
        for (int t = 0; t < T_; ++t) {
            float v = rowData[(b * T_ + t) * 4 + d];
            mn = fminf(mn, v);
            mx = fmaxf(mx, v);
        }
        float thres = THRESH_ * sscales[0];
        bounds[tid]      = mn - thres;            // lvals
        bounds[24 + tid] = mx + thres;            // uvals
    }
}

// ---------------------------------------------------------------------------
// Kernel 2: pack per-splat struct {x,y,z,m2 | rad+MARGIN,maskbits,0,0}  (32B)
// ---------------------------------------------------------------------------
__global__ __launch_bounds__(256)
void pack_means(const float* __restrict__ means,   // (N,3)
                const float* __restrict__ scales,  // (N,3)
                const float* __restrict__ bounds,  // lv[24] | uv[24]
                float*       __restrict__ pk)      // N*8
{
    int n = blockIdx.x * blockDim.x + threadIdx.x;
    if (n >= N_) return;
    float x = means[3 * n], y = means[3 * n + 1], z = means[3 * n + 2];
    float s0 = scales[3 * n], s1 = scales[3 * n + 1], s2 = scales[3 * n + 2];
    float radm = fmaxf(s0, fmaxf(s1, s2)) + MARGIN_;   // fold MARGIN here
    unsigned mask = 0;
    #pragma unroll
    for (int b = 0; b < B_; ++b) {
        bool in = x >= bounds[b * 3 + 0] && x <= bounds[24 + b * 3 + 0] &&
                  y >= bounds[b * 3 + 1] && y <= bounds[24 + b * 3 + 1] &&
                  z >= bounds[b * 3 + 2] && z <= bounds[24 + b * 3 + 2];
        mask |= (in ? 1u : 0u) << b;
    }
    float4 p0 = make_float4(x, y, z, x * x + y * y + z * z);
    float4 p1 = make_float4(radm, __uint_as_float(mask), 0.0f, 0.0f);
    ((float4*)pk)[n * 2 + 0] = p0;
    ((float4*)pk)[n * 2 + 1] = p1;
}

// ---------------------------------------------------------------------------
// Kernel 3: WMMA distance tiles + running min, 2 column tiles per iteration.
//   D = A(16x4) * B(4x16) + C :  A row = [-2x,-2y,-2z,1], B col = [mx,my,mz,m2],
//   C = r2 broadcast per row  ->  D = r2 + m2 - 2*cross = d2
// ---------------------------------------------------------------------------
__global__ __launch_bounds__(256)
void collide_min(const float*    __restrict__ rowData,
                 const unsigned* __restrict__ bcodes,
                 const float*    __restrict__ pk,
                 unsigned*       __restrict__ gmin)
{
    const int lane = threadIdx.x & 31;
    const int wave = threadIdx.x >> 5;
    const int wavesPerBlock = blockDim.x >> 5;
    const int half = lane >> 4;        // 0: K=0,1 / rows M..  1: K=2,3 / rows M+8
    const int l15  = lane & 15;

    for (int rt = wave; rt < ROW_TILES; rt += wavesPerBlock) {
        // ---- A matrix (2 VGPRs): lanes<16 hold K=0,1; lanes>=16 hold K=2,3
        int arow = rt * 16 + l15;
        float4 rd = ((const float4*)rowData)[arow];
        v2f a;
        a.x = half ? (-2.0f * rd.z) : (-2.0f * rd.x);
        a.y = half ? 1.0f           : (-2.0f * rd.y);

        // ---- C init: VGPR i holds r2 of row rt*16 + half*8 + i (all lanes in half)
        v8f cinit;
        #pragma unroll
        for (int i = 0; i < 8; ++i)
            cinit[i] = rowData[(rt * 16 + half * 8 + i) * 4 + 3];

        // ---- hoisted mask-bit selectors (loop-invariant per row tile)
        unsigned bc = bcodes[rt * 2 + half];
        unsigned bitsel[8];
        #pragma unroll
        for (int i = 0; i < 8; ++i)
            bitsel[i] = 1u << ((bc >> (4 * i)) & 0xFu);

        v8f runmin;
        #pragma unroll
        for (int i = 0; i < 8; ++i) runmin[i] = BIG_;

        for (int ct = blockIdx.x; ct < COL_PAIRS; ct += gridDim.x) {
            int n0 = (2 * ct)     * 16 + l15;          // this lane's columns
            int n1 = (2 * ct + 1) * 16 + l15;
            const float4* ppa = (const float4*)(pk + (size_t)n0 * 8);
            const float4* ppb = (const float4*)(pk + (size_t)n1 * 8);
            float4 p0a = ppa[0], p1a = ppa[1];
            float4 p0b = ppb[0], p1b = ppb[1];

            int ctn = ct + gridDim.x;                  // uniform prefetch of next stripe
            if (ctn < COL_PAIRS)
                __builtin_prefetch(pk + (size_t)(2 * ctn * 16 + l15) * 8, 0, 0);

            // B matrices (2 VGPRs each): lanes<16 rows K=0,1; lanes>=16 rows K=2,3
            v2f bma, bmb;
            bma.x = half ? p0a.z : p0a.x;
            bma.y = half ? p0a.w : p0a.y;
            bmb.x = half ? p0b.z : p0b.x;
            bmb.y = half ? p0b.w : p0b.y;

            v8f da = __builtin_amdgcn_wmma_f32_16x16x4_f32(
                false, a, false, bma, (short)0, cinit, false, false);
            v8f db = __builtin_amdgcn_wmma_f32_16x16x4_f32(
                false, a, false, bmb, (short)0, cinit, false, false);

            unsigned maska = __float_as_uint(p1a.y);
            unsigned maskb = __float_as_uint(p1b.y);
            #pragma unroll
            for (int i = 0; i < 8; ++i) {
                // single v_med3 clamp (no canonicalize), raw v_sqrt_f32 (TRANS)
                float dista = __builtin_amdgcn_sqrtf(
                                  __builtin_amdgcn_fmed3f(da[i], 0.0f, HUGE_));
                float distb = __builtin_amdgcn_sqrtf(
                                  __builtin_amdgcn_fmed3f(db[i], 0.0f, HUGE_));
                float resa  = dista - p1a.x;           // radm already includes MARGIN
                float resb  = distb - p1b.x;
                float va = (maska & bitsel[i]) ? resa : BIG_;
                float vb = (maskb & bitsel[i]) ? resb : BIG_;
                runmin[i] = fminf(runmin[i], fminf(va, vb));
            }
        }

        // ---- reduce across the 16 columns held in each half-wave, then atomic min
        #pragma unroll
        for (int i = 0; i < 8; ++i) {
            float v = runmin[i];
            #pragma unroll
            for (int off = 8; off >= 1; off >>= 1)
                v = fminf(v, __shfl_xor(v, off, 16));
            if (l15 == 0) {
                unsigned bits = __float_as_uint(v);
                unsigned key  = (bits & 0x80000000u) ? ~bits : (bits | 0x80000000u);
                atomicMin(&gmin[rt * 16 + half * 8 + i], key);
            }
        }
    }
}

// ---------------------------------------------------------------------------
// Kernel 4: decode keys, sum relu(-min)/ (B*T)
// ---------------------------------------------------------------------------
__global__ __launch_bounds__(256)
void finalize(const unsigned* __restrict__ gmin, float* __restrict__ out)
{
    __shared__ float sm[256];
    int tid = threadIdx.x;
    float s = 0.0f;
    for (int r = tid; r < ROWS; r += 256) {
        unsigned k = gmin[r];
        float f = (k & 0x80000000u) ? __uint_as_float(k ^ 0x80000000u)
                                    : __uint_as_float(~k);
        s += fmaxf(0.0f, -f);
    }
    sm[tid] = s;
    __syncthreads();
    #pragma unroll
    for (int off = 128; off >= 1; off >>= 1) {
        if (tid < off) sm[tid] += sm[tid + off];
        __syncthreads();
    }
    if (tid == 0) out[0] = sm[0] / (float)ROWS;
}

// ---------------------------------------------------------------------------
extern "C" void kernel_launch(void* const* d_in, const int* in_sizes, int n_in,
                              void* d_out, int out_size, void* d_ws, size_t ws_size,
                              hipStream_t stream)
{
    const float* outputs = (const float*)d_in[0];   // (B,T,3)
    const float* c2ws    = (const float*)d_in[1];   // (B,4,4)
    const float* sscales = (const float*)d_in[2];   // (B,)
    const float* means   = (const float*)d_in[3];   // (N,3)
    const float* scales  = (const float*)d_in[4];   // (N,3)
    float* out = (float*)d_out;

    char* ws = (char*)d_ws;
    float*    rowData = (float*)   (ws + 0);
    float*    bounds  = (float*)   (ws + 12800);
    unsigned* bcodes  = (unsigned*)(ws + 12992);
    unsigned* gmin    = (unsigned*)(ws + 13392);
    float*    pk      = (float*)   (ws + 16592);

    prep_rows<<<1, 1024, 0, stream>>>(outputs, c2ws, sscales,
                                      rowData, bounds, bcodes, gmin);
    pack_means<<<(N_ + 255) / 256, 256, 0, stream>>>(means, scales, bounds, pk);
    collide_min<<<256, 256, 0, stream>>>(rowData, bcodes, pk, gmin);
    finalize<<<1, 256, 0, stream>>>(gmin, out);
}